// LIFTWrapper_10591389352525
// MI455X (gfx1250) — compile-verified
//
#include <hip/hip_runtime.h>

// ---------------------------------------------------------------------------
// LIF spiking-neuron scan (snnTorch Leaky forward), MI455X / gfx1250.
//
//   mem_t = 0.9*mem_{t-1} + x_t - spk_{t-1}        (thresh = 1, subtract reset)
//   spk_t = (mem_t > 1) ? 1 : 0
//
// x: [T=16, N=2,097,152] f32 -> spikes same shape.
// Memory-bound: 268 MB total traffic -> ~11.5 us floor @ 23.3 TB/s. Working
// set (256 MiB, zero reuse) exceeds the 192 MB L2, so both streams are marked
// non-temporal. HBM->LDS via the CDNA5 async DMA path
// (GLOBAL_LOAD_ASYNC_TO_LDS_B128 / ASYNCcnt), double-buffered per t-step;
// per-neuron scan state stays in VGPRs across all 16 steps.
// ---------------------------------------------------------------------------

#define LIF_BETA 0.9f

constexpr int kT   = 16;   // time steps (fixed by reference shape)
constexpr int kTPB = 256;  // 8 wave32 per block; each thread owns 4 neurons

// gfx12-style CPol: TH in bits [2:0]; TH_NT = 1 (non-temporal).
constexpr int kCpolNT = 1;

// Native clang vectors (HIP's float4 is a struct and unusable with the
// nontemporal builtins).
typedef float v4f __attribute__((ext_vector_type(4)));
typedef int   v4i __attribute__((ext_vector_type(4)));

// The async builtins take typed v4i pointers in explicit address spaces:
//   __builtin_amdgcn_global_load_async_to_lds_b128(v4i AS1*, v4i AS3*, Ii, Ii)
typedef __attribute__((address_space(1))) v4i gbl_v4i;   // global (64-bit ptr)
typedef __attribute__((address_space(3))) v4i lds_v4i;   // LDS    (32-bit ptr)

// Issue one 16-byte async global->LDS copy for this lane (ASYNCcnt-tracked),
// non-temporal so the once-read input stream is evict-first in cache.
__device__ __forceinline__ void async_copy_b128(const float* g, float* l) {
#if __has_builtin(__builtin_amdgcn_global_load_async_to_lds_b128)
  __builtin_amdgcn_global_load_async_to_lds_b128(
      (gbl_v4i*)(unsigned long long)g,
      (lds_v4i*)(unsigned int)(unsigned long long)l,
      /*imm offset=*/0, /*cpol=*/kCpolNT);
#else
  unsigned int lds_addr = (unsigned int)(unsigned long long)l;
  asm volatile("global_load_async_to_lds_b128 %0, %1, off th:TH_LOAD_NT"
               :
               : "v"(lds_addr), "v"(g)
               : "memory");
#endif
  asm volatile("" ::: "memory");  // keep LDS reads from hoisting above issue
}

template <int CNT>
__device__ __forceinline__ void wait_asynccnt() {
#if __has_builtin(__builtin_amdgcn_s_wait_asynccnt)
  __builtin_amdgcn_s_wait_asynccnt(CNT);
#else
  asm volatile("s_wait_asynccnt %0" : : "n"(CNT) : "memory");
#endif
  asm volatile("" ::: "memory");  // compiler fence: ds_load stays below wait
}

__global__ void __launch_bounds__(kTPB)
lif_scan_async(const float* __restrict__ x, float* __restrict__ out, int n4) {
  // Double-buffered stage: 2 x 4 KiB. Each lane touches only its own 16 B,
  // so no cross-wave sharing and no barriers are needed.
  __shared__ alignas(16) float tile[2][kTPB * 4];

  const int tid = threadIdx.x;
  const long long q = (long long)blockIdx.x * kTPB + tid;  // float4 slot
  if (q >= n4) return;

  const long long N    = 4LL * n4;     // elements per time slice
  const float*    gsrc = x + 4 * q;    // this lane's address in slice t=0
  float* l0 = &tile[0][tid * 4];
  float* l1 = &tile[1][tid * 4];

  // Prefetch t = 0 into buffer 0.
  async_copy_b128(gsrc, l0);

  // Neuron state lives in VGPRs for the whole scan.
  float m0 = 0.f, m1 = 0.f, m2 = 0.f, m3 = 0.f;
  float s0 = 0.f, s1 = 0.f, s2 = 0.f, s3 = 0.f;

#pragma unroll
  for (int t = 0; t < kT; ++t) {
    float* lc = (t & 1) ? l1 : l0;   // buffer holding slice t
    float* ln = (t & 1) ? l0 : l1;   // buffer for slice t+1
    if (t + 1 < kT) {
      // Issue next slice, then retire the oldest outstanding copy (slice t):
      // async loads complete in order, so ASYNCcnt <= 1 ==> slice t is in LDS.
      async_copy_b128(gsrc + (long long)(t + 1) * N, ln);
      wait_asynccnt<1>();
    } else {
      wait_asynccnt<0>();
    }

    const v4f xv = *reinterpret_cast<const v4f*>(lc);  // ds_load_b128

    // mem = beta*mem + x - spk  (thresh == 1)
    m0 = __builtin_fmaf(LIF_BETA, m0, xv.x) - s0;
    m1 = __builtin_fmaf(LIF_BETA, m1, xv.y) - s1;
    m2 = __builtin_fmaf(LIF_BETA, m2, xv.z) - s2;
    m3 = __builtin_fmaf(LIF_BETA, m3, xv.w) - s3;
    // spk = Heaviside(mem - 1)
    s0 = (m0 > 1.0f) ? 1.0f : 0.0f;
    s1 = (m1 > 1.0f) ? 1.0f : 0.0f;
    s2 = (m2 > 1.0f) ? 1.0f : 0.0f;
    s3 = (m3 > 1.0f) ? 1.0f : 0.0f;

    v4f sp;
    sp.x = s0; sp.y = s1; sp.z = s2; sp.w = s3;
    // Non-temporal b128 store: output is written once, never re-read here;
    // keep 128 MiB of spikes from dirtying L2 against the input stream.
    __builtin_nontemporal_store(
        sp, reinterpret_cast<v4f*>(out + (long long)t * N + 4 * q));
  }
}

extern "C" void kernel_launch(void* const* d_in, const int* in_sizes, int n_in,
                              void* d_out, int out_size, void* d_ws,
                              size_t ws_size, hipStream_t stream) {
  (void)n_in; (void)out_size; (void)d_ws; (void)ws_size;

  const float* x   = (const float*)d_in[0];
  float*       out = (float*)d_out;

  const long long total = in_sizes[0];          // T * N
  const int N  = (int)(total / kT);             // 2,097,152 (divisible by 4)
  const int n4 = N / 4;                         // float4 slots per slice
  const int blocks = (n4 + kTPB - 1) / kTPB;    // 2048 blocks

  lif_scan_async<<<blocks, kTPB, 0, stream>>>(x, out, n4);
}